// GradientFeatureMasking_38903813767349
// MI455X (gfx1250) — compile-verified
//
#include <hip/hip_runtime.h>

// ---------------------------------------------------------------------------
// GradientFeatureMasking forward == identity on x (stop_gradient only alters
// the backward pass). d_out = copy of d_in[0] (32768*1024 fp32, 128 MiB).
// Pure HBM-bandwidth problem: ~268 MB traffic -> ~11.5 us at 23.3 TB/s.
// Streamed through the CDNA5 async global<->LDS engine (ASYNCcnt path,
// GLOBAL_LOAD_ASYNC_TO_LDS_B128 / GLOBAL_STORE_ASYNC_FROM_LDS_B128).
//
// Builtin signature (revealed by round-1 diagnostic):
//   void __builtin_amdgcn_global_load_async_to_lds_b128(
//       int4 addrspace(1)* gaddr, int4 addrspace(3)* lds, imm offset, imm cpol)
// ---------------------------------------------------------------------------

typedef int v4i __attribute__((vector_size(16)));
typedef __attribute__((address_space(1))) v4i as1_v4i;
typedef __attribute__((address_space(3))) v4i as3_v4i;

#if defined(__has_builtin)
#  if __has_builtin(__builtin_amdgcn_global_load_async_to_lds_b128) && \
      __has_builtin(__builtin_amdgcn_global_store_async_from_lds_b128)
#    define USE_ASYNC_BUILTINS 1
#  endif
#  if __has_builtin(__builtin_amdgcn_s_wait_asynccnt)
#    define USE_WAIT_BUILTIN 1
#  endif
#endif

__device__ __forceinline__ void wait_asynccnt0() {
#if defined(USE_WAIT_BUILTIN)
  __builtin_amdgcn_s_wait_asynccnt(0);
#else
  asm volatile("s_wait_asynccnt 0" ::: "memory");
#endif
}

#define COPY_BLOCK 256

__global__ __launch_bounds__(COPY_BLOCK) void
GradientFeatureMasking_copy_async(const float4* __restrict__ src,
                                  float4* __restrict__ dst,
                                  long long n4) {
  // One 16-byte LDS slot per lane; slots are lane-private, so no barriers are
  // needed — only ASYNCcnt waits. Loads return done in order and stores in
  // order, but loads vs stores are unordered, so with a single buffer only
  // asynccnt==0 waits are safe.
  __shared__ alignas(16) float4 lbuf[COPY_BLOCK];

  const int tid = threadIdx.x;
  long long i = (long long)blockIdx.x * COPY_BLOCK + tid;
  const long long stride = (long long)gridDim.x * COPY_BLOCK;

  as3_v4i* lslot = (as3_v4i*)&lbuf[tid];

  for (; i < n4; i += stride) {
#if defined(USE_ASYNC_BUILTINS)
    // global -> LDS (b128, ASYNCcnt++)
    __builtin_amdgcn_global_load_async_to_lds_b128(
        (as1_v4i*)(src + i), lslot, /*offset=*/0, /*cpol=*/0);
    wait_asynccnt0();  // load landed in LDS
    // LDS -> global (b128, ASYNCcnt++)
    __builtin_amdgcn_global_store_async_from_lds_b128(
        (as1_v4i*)(dst + i), lslot, /*offset=*/0, /*cpol=*/0);
    wait_asynccnt0();  // slot reusable next iteration
#else
    // Inline-asm fallback per CDNA5 ISA (GV mode).
    unsigned laddr = (unsigned)(unsigned long long)lslot;  // LDS byte address
    const void* sp = (const void*)(src + i);
    void* dp = (void*)(dst + i);
    asm volatile(
        "global_load_async_to_lds_b128 %0, %1, off\n\t"
        "s_wait_asynccnt 0\n\t"
        "global_store_async_from_lds_b128 %2, %0, off\n\t"
        "s_wait_asynccnt 0"
        :
        : "v"(laddr), "v"(sp), "v"(dp)
        : "memory");
#endif
  }
}

// Scalar tail for n not divisible by 4 (not hit for 32768*1024, kept for
// correctness on any size).
__global__ void GradientFeatureMasking_copy_tail(const float* __restrict__ src,
                                                 float* __restrict__ dst,
                                                 long long start, long long n) {
  long long i = start + (long long)blockIdx.x * blockDim.x + threadIdx.x;
  if (i < n) dst[i] = src[i];
}

extern "C" void kernel_launch(void* const* d_in, const int* in_sizes, int n_in,
                              void* d_out, int out_size, void* d_ws, size_t ws_size,
                              hipStream_t stream) {
  (void)n_in; (void)d_ws; (void)ws_size; (void)out_size;

  const float* x = (const float*)d_in[0];   // x : [32768, 1024] fp32
  // d_in[1] (rand_u) and d_in[2] (feat_choice) do not affect the forward value.
  float* out = (float*)d_out;

  const long long n  = (long long)in_sizes[0];
  const long long n4 = n >> 2;

  if (n4 > 0) {
    long long blocks_needed = (n4 + COPY_BLOCK - 1) / COPY_BLOCK;
    int grid = (int)(blocks_needed < 16384 ? blocks_needed : 16384);
    // 16384 blocks * 256 threads = 4.19M threads (131k wave32s): each does two
    // b128 async transfers; plenty of waves in flight to hide the asynccnt
    // round-trips and keep all HBM channels saturated.
    GradientFeatureMasking_copy_async<<<grid, COPY_BLOCK, 0, stream>>>(
        (const float4*)x, (float4*)out, n4);
  }

  const long long rem = n - (n4 << 2);
  if (rem > 0) {
    GradientFeatureMasking_copy_tail<<<1, 256, 0, stream>>>(x, out, n4 << 2, n);
  }
}